// Trend_Model_52226802319619
// MI455X (gfx1250) — compile-verified
//
#include <hip/hip_runtime.h>

// Holt double-exponential smoothing, reformulated as a chunked linear-scan
// where each 16-timestep chunk is a 16x16 matmul executed with
// V_WMMA_F32_16X16X4_F32 (4 chained K=4 WMMAs per chunk).
//
//   state_t = M * state_{t-1} + v * x_t,   state = (s, b)
//   M = [[1-a, 1-a], [-a*B, B*(1-a)+(1-B)]],  v = (a, a*B)

#define HALPHA 0.8f
#define HBETA  0.2f
#define M00 (1.0f - HALPHA)
#define M01 (1.0f - HALPHA)
#define M10 (-(HALPHA * HBETA))
#define M11 (HBETA * (1.0f - HALPHA) + (1.0f - HBETA))

typedef float v2f __attribute__((ext_vector_type(2)));
typedef float v8f __attribute__((ext_vector_type(8)));

// (M^d * v), component comp (0 = s-coef, 1 = b-coef)
__device__ __forceinline__ float g_comp(int d, int comp) {
  float p = HALPHA, q = HALPHA * HBETA;
  for (int i = 0; i < d; ++i) {
    float np = M00 * p + M01 * q;
    float nq = M10 * p + M11 * q;
    p = np; q = nq;
  }
  return comp ? q : p;
}

__global__ __launch_bounds__(256) void holt_wmma_kernel(
    const float* __restrict__ x, float* __restrict__ out) {
  constexpr int T = 4096, C = 512;
  constexpr int NCH = 255;               // chunks of 16 covering t = 16..4095

  const int tid  = blockIdx.x * blockDim.x + threadIdx.x;
  const int lane = threadIdx.x & 31;
  const int wave = tid >> 5;             // 0..1023
  const int b    = wave >> 5;            // 0..31  (batch)
  const int c0   = (wave & 31) << 4;     // channel group base
  const int n    = lane & 15;            // channel within group (N dim)
  const int half = lane >> 4;            // 0 or 1 (half-wave)

  // ---------------- one-time per-lane coefficient setup ----------------
  v2f afrag[4];              // A fragments: W[m][k], m = n, k = 4kk + r + 2*half
  v2f ufrag[4];              // u[k] = (M^{15-k} v)_1 at this lane's K slots
#pragma unroll
  for (int kk = 0; kk < 4; ++kk) {
    int k0 = 4 * kk + half * 2;
    float w0 = (n >= k0)     ? g_comp(n - k0, 0)     : 0.0f;
    float w1 = (n >= k0 + 1) ? g_comp(n - (k0 + 1), 0) : 0.0f;
    afrag[kk].x = w0; afrag[kk].y = w1;
    ufrag[kk].x = g_comp(15 - k0, 1);
    ufrag[kk].y = g_comp(15 - (k0 + 1), 1);
  }
  // accumulator-init coefficients: row0 of M^{i+1}, i = g + 8*half
  float arow[8], crow[8];
  {
    float p = 1.0f, q = 0.0f;            // row0 of M^0
#pragma unroll
    for (int e = 1; e <= 16; ++e) {
      float np = p * M00 + q * M10;
      float nq = p * M01 + q * M11;
      p = np; q = nq;                    // row0 of M^e
      int g = (e - 1) - half * 8;
      if (g >= 0 && g < 8) { arow[g] = p; crow[g] = q; }
    }
  }
  // row1 of M^16 (carry coefficients for b)
  float m16_10, m16_11;
  {
    float p = 0.0f, q = 1.0f;
#pragma unroll
    for (int e = 0; e < 16; ++e) {
      float np = p * M00 + q * M10;
      float nq = p * M01 + q * M11;
      p = np; q = nq;
    }
    m16_10 = p; m16_11 = q;
  }

  // ---------------- scalar warm-up: t = 0..15 ----------------
  const size_t base = ((size_t)b * T) * C + (size_t)(c0 + n);
  float s  = x[base];                    // s0 = x[0]
  if (half == 0) out[base] = s;
  float bt = x[base + C] - s;            // b0 = x[1] - x[0]
#pragma unroll 1
  for (int t = 1; t <= 15; ++t) {
    float xi = x[base + (size_t)t * C];
    float sn = HALPHA * xi + (1.0f - HALPHA) * (s + bt);
    bt = HBETA * (sn - s) + (1.0f - HBETA) * bt;
    s = sn;
    if (half == 0) out[base + (size_t)t * C] = s;
  }

  // ---------------- WMMA chunk loop: t = 16..4095, 255 chunks ----------------
  const float* xp = x   + base + (size_t)16 * C + (size_t)half * 2 * C;
  float*       op = out + base + (size_t)16 * C + (size_t)half * 8 * C;

  v2f cur[4], nxt[4];
#pragma unroll
  for (int kk = 0; kk < 4; ++kk) {       // first chunk's B fragments
    cur[kk].x = xp[(size_t)(4 * kk + 0) * C];
    cur[kk].y = xp[(size_t)(4 * kk + 1) * C];
  }

  for (int ch = 0; ch < NCH; ++ch) {
    // prefetch next chunk into registers (double buffer)
    const float* xpn = xp + (size_t)(ch + 1) * 16 * C;
    if (ch + 1 < NCH) {
#pragma unroll
      for (int kk = 0; kk < 4; ++kk) {
        nxt[kk].x = xpn[(size_t)(4 * kk + 0) * C];
        nxt[kk].y = xpn[(size_t)(4 * kk + 1) * C];
      }
    }
    // warm L2 a few chunks ahead (global_prefetch_b8)
    if (ch + 6 < NCH) __builtin_prefetch(xp + (size_t)(ch + 6) * 16 * C, 0, 0);

    // C init with incoming-state contribution, then 4x K=4 f32 WMMA
    v8f acc;
#pragma unroll
    for (int g = 0; g < 8; ++g) acc[g] = arow[g] * s + crow[g] * bt;
#pragma unroll
    for (int kk = 0; kk < 4; ++kk) {
      acc = __builtin_amdgcn_wmma_f32_16x16x4_f32(
          false, afrag[kk], false, cur[kk], (short)0, acc, false, false);
    }

    // store 16x16 output tile (each lane: 8 rows, coalesced per half-row)
    float* o = op + (size_t)ch * 16 * C;
#pragma unroll
    for (int g = 0; g < 8; ++g) o[(size_t)g * C] = acc[g];

    // carry update for next chunk:
    //   b' = (M^16)_10 s + (M^16)_11 b + u . x_chunk   (cross-half reduce)
    float dot = 0.0f;
#pragma unroll
    for (int kk = 0; kk < 4; ++kk)
      dot += ufrag[kk].x * cur[kk].x + ufrag[kk].y * cur[kk].y;
    dot += __shfl_xor(dot, 16, 32);                 // ds_swizzle SWAPX16
    float bnew = m16_10 * s + m16_11 * bt + dot;
    //   s' = D[15][n], lives in acc[7] of lanes 16..31 -> broadcast
    float snew = __shfl(acc[7], n + 16, 32);        // ds_bpermute
    s = snew; bt = bnew;

#pragma unroll
    for (int kk = 0; kk < 4; ++kk) cur[kk] = nxt[kk];
  }
}

extern "C" void kernel_launch(void* const* d_in, const int* in_sizes, int n_in,
                              void* d_out, int out_size, void* d_ws, size_t ws_size,
                              hipStream_t stream) {
  (void)in_sizes; (void)n_in; (void)out_size; (void)d_ws; (void)ws_size;
  const float* x = (const float*)d_in[0];
  float* out = (float*)d_out;
  // 1024 waves = 32 batches * 32 channel-groups; 8 waves per block
  dim3 grid(128), block(256);
  hipLaunchKernelGGL(holt_wmma_kernel, grid, block, 0, stream, x, out);
}